// AudioTokenizer_17927193493858
// MI455X (gfx1250) — compile-verified
//
#include <hip/hip_runtime.h>

// VQ codebook search: z (32768 x 256) vs codebook (8192 x 256), fp32.
// score(n,k) = ||c_k||^2 - 2 z_n.c_k  (||z||^2 constant per row, dropped for argmin)
// Outputs (concat flat): z_q (32768*256 f32), token_ids (32768, as f32), loss (1 f32).

#define NROWS     32768
#define KTOK      8192
#define DDIM      256
#define KTILES    (KTOK / 16)     // 512
#define DSTEPS    (DDIM / 4)      // 64 wmma_f32_16x16x4 per 16x16 tile
#define LDSROW    260             // 256 + 4 dword pad -> conflict-free ds_load_b64
#define LDSBUF    (16 * LDSROW)   // floats per tile buffer
#define ND        (NROWS * DDIM)

typedef __attribute__((ext_vector_type(2))) float v2f;
typedef __attribute__((ext_vector_type(8))) float v8f;

// ---------------- kernel 0: codebook squared norms -> ws[0..8191] ----------------
__global__ __launch_bounds__(256) void vq_cnorm_kernel(const float* __restrict__ cb,
                                                       float* __restrict__ cnorm) {
    const int wid  = threadIdx.x >> 5;
    const int lane = threadIdx.x & 31;
    const int row  = blockIdx.x * 8 + wid;           // grid = 1024 blocks
    const float* r = cb + row * DDIM;
    float s = 0.f;
#pragma unroll
    for (int j = 0; j < 8; ++j) { float v = r[lane + 32 * j]; s = fmaf(v, v, s); }
#pragma unroll
    for (int m = 16; m >= 1; m >>= 1) s += __shfl_xor(s, m, 32);
    if (lane == 0) cnorm[row] = s;
}

// ---------------- kernel 1: fused GEMM + argmin + gather + loss partials --------
__global__ __launch_bounds__(256, 1) void vq_search_kernel(const float* __restrict__ z,
                                                           const float* __restrict__ cb,
                                                           const float* __restrict__ cnorm,
                                                           float* __restrict__ out,
                                                           float* __restrict__ partials) {
    __shared__ float cb_s[2][LDSBUF];        // double-buffered codebook tile (padded)
    __shared__ int   best_s[8 * 16];         // per-wave winning token per row
    __shared__ float red_s[256];             // loss reduction

    const int tid  = threadIdx.x;
    const int wid  = tid >> 5;               // wave 0..7
    const int lane = tid & 31;
    const int l16  = lane & 15;              // N (codebook col) / A-row selector
    const int half = lane >> 4;              // 0|1
    const int off  = half * 2;               // dim sub-offset within K-group of 4

    const int rowBase = blockIdx.x * 128 + wid * 16;  // grid = 256 blocks
    const int myArow  = rowBase + l16;

    // ---- A tile resident in registers, exact 16x4-f32 fragment layout ----
    float2 a[DSTEPS];
    {
        const float* zr = z + myArow * DDIM;
#pragma unroll
        for (int t = 0; t < DSTEPS; ++t)
            a[t] = *(const float2*)(zr + 4 * t + off);
    }

    // ---- async global->LDS tile copy (4 x b128 per thread per tile) ----
    // thread covers row srow, float4 cols sc4 + {0,16,32,48}; both the global
    // and LDS strides of the 4 chunks are 256 B, so one vaddr/ldsaddr pair plus
    // immediate offsets 0/256/512/768 describes all four transfers.
    const int srow = tid >> 4;               // 0..15
    const int sc4  = tid & 15;               // 0..15 (float4 col)
    const unsigned lds_mine =
        (unsigned)(size_t)&cb_s[0][0] + (unsigned)(srow * (LDSROW * 4) + sc4 * 16);
    const unsigned long long cb64 = (unsigned long long)(size_t)cb;
    const unsigned gmine = (unsigned)(srow * 1024 + sc4 * 16);  // bytes within tile

    // preload tile 0 into buffer 0
    {
        unsigned go = gmine;
        unsigned la = lds_mine;
        asm volatile(
            "global_load_async_to_lds_b128 %0, %1, %2\n\t"
            "global_load_async_to_lds_b128 %0, %1, %2 offset:256\n\t"
            "global_load_async_to_lds_b128 %0, %1, %2 offset:512\n\t"
            "global_load_async_to_lds_b128 %0, %1, %2 offset:768"
            :: "v"(la), "v"(go), "s"(cb64) : "memory");
        asm volatile("s_wait_asynccnt 0x0" ::: "memory");
    }
    __syncthreads();

    float bestv[8];
    int   besti[8];
#pragma unroll
    for (int r = 0; r < 8; ++r) { bestv[r] = 3.4e38f; besti[r] = 0; }

    for (int kt = 0; kt < KTILES; ++kt) {
        const int cur = kt & 1;
        // kick off async copy of next tile into the other buffer
        if (kt + 1 < KTILES) {
            unsigned go = gmine + (unsigned)(kt + 1) * 16384u;       // 16 rows * 1 KB
            unsigned la = lds_mine + (unsigned)(cur ^ 1) * (LDSBUF * 4u);
            asm volatile(
                "global_load_async_to_lds_b128 %0, %1, %2\n\t"
                "global_load_async_to_lds_b128 %0, %1, %2 offset:256\n\t"
                "global_load_async_to_lds_b128 %0, %1, %2 offset:512\n\t"
                "global_load_async_to_lds_b128 %0, %1, %2 offset:768"
                :: "v"(la), "v"(go), "s"(cb64) : "memory");
        }
        const float cn = cnorm[kt * 16 + l16];

        // 6 rotating accumulators -> accumulator RAW distance 6, beyond the
        // S_DELAY_ALU window; 48 VGPRs leaves room to batch the DS loads.
        v8f acc[6];
#pragma unroll
        for (int q = 0; q < 6; ++q) acc[q] = (v8f){};
        const float* bb = &cb_s[cur][l16 * LDSROW + off];
#pragma unroll
        for (int t = 0; t < DSTEPS; ++t) {
            float2 bv = *(const float2*)(bb + 4 * t);
            v2f A = {a[t].x, a[t].y};
            v2f B = {bv.x, bv.y};
            acc[t % 6] = __builtin_amdgcn_wmma_f32_16x16x4_f32(false, A, false, B,
                                                               (short)0, acc[t % 6],
                                                               false, false);
        }
        const int k = kt * 16 + l16;
#pragma unroll
        for (int r = 0; r < 8; ++r) {
            float sum = ((acc[0][r] + acc[1][r]) + (acc[2][r] + acc[3][r])) +
                        (acc[4][r] + acc[5][r]);
            float s = fmaf(-2.0f, sum, cn);
            if (s < bestv[r]) { bestv[r] = s; besti[r] = k; }
        }
        // my async writes into buf[cur^1] must land before the barrier publishes it
        asm volatile("s_wait_asynccnt 0x0" ::: "memory");
        __syncthreads();
    }

    // ---- cross-lane (16-wide) argmin reduction; ties -> lowest index ----
#pragma unroll
    for (int r = 0; r < 8; ++r) {
        float v = bestv[r]; int i = besti[r];
#pragma unroll
        for (int m = 1; m <= 8; m <<= 1) {
            float ov = __shfl_xor(v, m, 32);
            int   oi = __shfl_xor(i, m, 32);
            if (ov < v || (ov == v && oi < i)) { v = ov; i = oi; }
        }
        bestv[r] = v; besti[r] = i;
    }
    if (l16 == 0) {
#pragma unroll
        for (int r = 0; r < 8; ++r) best_s[wid * 16 + half * 8 + r] = besti[r];
    }
    __syncthreads();

    // ---- gather z_q, write output, accumulate commitment-loss partial ----
    const int    myk = best_s[wid * 16 + l16];
    const float* cr  = cb + myk * DDIM;
    float*       orw = out + myArow * DDIM;
    float ls = 0.f;
#pragma unroll
    for (int t = 0; t < DSTEPS; ++t) {
        float2 c = *(const float2*)(cr + 4 * t + off);
        *(float2*)(orw + 4 * t + off) = c;
        float d0 = c.x - a[t].x, d1 = c.y - a[t].y;
        ls = fmaf(d0, d0, fmaf(d1, d1, ls));
    }
    if (half == 0)
        out[ND + rowBase + l16] = (float)best_s[wid * 16 + l16];

    // ---- block loss reduction -> partials[block] ----
    red_s[tid] = ls;
    __syncthreads();
#pragma unroll
    for (int s = 128; s >= 1; s >>= 1) {
        if (tid < s) red_s[tid] += red_s[tid + s];
        __syncthreads();
    }
    if (tid == 0) partials[blockIdx.x] = red_s[0];
}

// ---------------- kernel 2: finalize loss ----------------
__global__ __launch_bounds__(256) void vq_loss_kernel(const float* __restrict__ partials,
                                                      float* __restrict__ out) {
    __shared__ float red_s[256];
    red_s[threadIdx.x] = partials[threadIdx.x];   // 256 partials, 1 block
    __syncthreads();
#pragma unroll
    for (int s = 128; s >= 1; s >>= 1) {
        if (threadIdx.x < s) red_s[threadIdx.x] += red_s[threadIdx.x + s];
        __syncthreads();
    }
    if (threadIdx.x == 0)
        out[ND + NROWS] = 0.25f * red_s[0] / (float)ND;
}

extern "C" void kernel_launch(void* const* d_in, const int* in_sizes, int n_in,
                              void* d_out, int out_size, void* d_ws, size_t ws_size,
                              hipStream_t stream) {
    const float* z  = (const float*)d_in[0];   // (32,1024,256) f32
    const float* cb = (const float*)d_in[1];   // (8192,256) f32
    float* out      = (float*)d_out;           // z_q | token_ids | loss
    float* cnorm    = (float*)d_ws;            // [0..8191]
    float* partials = cnorm + KTOK;            // [8192..8447]

    vq_cnorm_kernel<<<KTOK / 8, 256, 0, stream>>>(cb, cnorm);
    vq_search_kernel<<<NROWS / 128, 256, 0, stream>>>(z, cb, cnorm, out, partials);
    vq_loss_kernel<<<1, 256, 0, stream>>>(partials, out);
}